// TSALoss_21827023799003
// MI455X (gfx1250) — compile-verified
//
#include <hip/hip_runtime.h>

// ---------------------------------------------------------------------------
// TSA loss for MI455X (gfx1250, wave32).
//   B=16384 points, D=16, K=25 neighbors, P=1 (rank-1 projections).
//   loss = mean((out-tgt)^2) + 0.1 * mean_b( 2 - 2*(uz_b . ux_b)^2 )
// where uz/ux are top eigenvectors of the 16x16 neighbor covariances.
// Gram matrix and covariances computed with V_WMMA_F32_16X16X4_F32.
// Workspace: [0]=recon accum, [1]=tsa accum, then 16384*25 int neighbor ids.
// ---------------------------------------------------------------------------

#define B_PTS   16384
#define DIMS    16
#define KNN     25
#define NWAVE   8
#define NCOL_CH (NWAVE * 16)          // 128 candidate columns per chunk
#define NCHUNK  (B_PTS / NCOL_CH)     // 128 chunks

typedef __attribute__((ext_vector_type(2))) float v2f;
typedef __attribute__((ext_vector_type(8))) float v8f;

__device__ __forceinline__ v8f wmma_f32_4(v2f a, v2f b, v8f c) {
  // D = A(16x4,f32) * B(4x16,f32) + C(16x16,f32)
  return __builtin_amdgcn_wmma_f32_16x16x4_f32(false, a, false, b,
                                               (short)0, c, false, false);
}

// ---------------------------------------------------------------------------
__global__ void zero_accum_kernel(float* ws) {
  if (threadIdx.x < 2) ws[threadIdx.x] = 0.0f;
}

// ---------------------------------------------------------------------------
__global__ void recon_kernel(const float* __restrict__ o,
                             const float* __restrict__ t,
                             float* __restrict__ acc, int n) {
  int idx    = blockIdx.x * blockDim.x + threadIdx.x;
  int stride = gridDim.x * blockDim.x;
  float s = 0.0f;
  for (int i = idx; i < n; i += stride) {
    float d = o[i] - t[i];
    s += d * d;
  }
  #pragma unroll
  for (int m = 1; m < 32; m <<= 1) s += __shfl_xor(s, m);
  if ((threadIdx.x & 31) == 0) atomicAdd(acc, s);
}

// ---------------------------------------------------------------------------
// One block = 16 query rows. 8 waves each produce one 16x16 distance tile per
// chunk via 4x V_WMMA_F32_16X16X4_F32 (K=16 Gram) fused with
// d2 = |q|^2 + |c|^2 - 2 q.c, staged in LDS; 16 selector threads maintain
// per-row sorted top-25 (self excluded) with a threshold gate.
__global__ __launch_bounds__(256) void knn_kernel(const float* __restrict__ raw,
                                                  int* __restrict__ nbr) {
  __shared__ float sQ[16][DIMS];
  __shared__ float sqQ[16];
  __shared__ float sDist[16][NCOL_CH];
  __shared__ float topD[16][KNN];
  __shared__ int   topJ[16][KNN];

  const int tid  = threadIdx.x;
  const int wav  = tid >> 5;
  const int lane = tid & 31;
  const int mn   = lane & 15;   // row (A) / col (B) index within tile
  const int lh   = lane >> 4;   // wave half selects K sub-pair
  const int i0   = blockIdx.x * 16;

  { // stage query block: 256 threads cover 16x16
    int m = tid >> 4, d = tid & 15;
    sQ[m][d] = raw[(size_t)(i0 + m) * DIMS + d];
  }
  __syncthreads();
  if (tid < 16) {
    float s = 0.0f;
    #pragma unroll
    for (int d = 0; d < DIMS; ++d) { float x = sQ[tid][d]; s += x * x; }
    sqQ[tid] = s;
  }
  __syncthreads();

  float worst = 3.4e38f;
  int   cnt   = 0;

  for (int chunk = 0; chunk < NCHUNK; ++chunk) {
    const int j0 = (chunk * NWAVE + wav) * 16;
    const float* cp = raw + (size_t)(j0 + mn) * DIMS;

    // A: lane holds Xq[mn][k], B: lane holds Xc[j0+mn][k]; k = 4*kc + 2*lh + {0,1}
    v2f a[4], bb[4];
    float s2 = 0.0f;
    #pragma unroll
    for (int kc = 0; kc < 4; ++kc) {
      const int k = 4 * kc + 2 * lh;
      a[kc].x = sQ[mn][k];
      a[kc].y = sQ[mn][k + 1];
      float b0 = cp[k], b1 = cp[k + 1];
      bb[kc].x = b0; bb[kc].y = b1;
      s2 += b0 * b0 + b1 * b1;
    }
    const float sqC = s2 + __shfl_xor(s2, 16);   // |Xc[j0+mn]|^2

    v8f accv = {};
    #pragma unroll
    for (int kc = 0; kc < 4; ++kc)
      accv = wmma_f32_4(a[kc], bb[kc], accv);    // Gram tile S[m][n]

    #pragma unroll
    for (int v = 0; v < 8; ++v) {
      const int m = v + 8 * lh;                  // D layout: lane=n, vgpr->m
      sDist[m][(wav << 4) + mn] = sqQ[m] + sqC - 2.0f * accv[v];
    }
    __syncthreads();

    if (tid < 16) {                              // one selector thread per row
      const int row = tid;
      const int jb  = chunk * NCOL_CH;
      for (int c = 0; c < NCOL_CH; ++c) {
        const int j = jb + c;
        if (j == i0 + row) continue;             // exclude self
        const float d = sDist[row][c];
        if (d >= worst) continue;
        int pos;
        if (cnt < KNN) pos = cnt++;
        else           pos = KNN - 1;
        while (pos > 0 && topD[row][pos - 1] > d) {
          topD[row][pos] = topD[row][pos - 1];
          topJ[row][pos] = topJ[row][pos - 1];
          --pos;
        }
        topD[row][pos] = d;
        topJ[row][pos] = j;
        if (cnt == KNN) worst = topD[row][KNN - 1];
      }
    }
    __syncthreads();                             // protect sDist for next chunk
  }

  if (tid < 16) {
    #pragma unroll
    for (int k = 0; k < KNN; ++k)
      nbr[(size_t)(i0 + tid) * KNN + k] = topJ[tid][k];
  }
}

// ---------------------------------------------------------------------------
// One wave per point: gather 25 neighbors, center, cov = c^T c via 7x f32 WMMA
// (K padded 25->28 with zero rows; A and B operands are the same registers),
// power-iterate the 16x16 cov in LDS for the top eigenvector of latent and raw,
// accumulate 2 - 2*(uz.ux)^2.
__global__ __launch_bounds__(128) void tsa_kernel(const float* __restrict__ latent,
                                                  const float* __restrict__ raw,
                                                  const int* __restrict__ nbr,
                                                  float* __restrict__ accTsa) {
  __shared__ float sCov[4][16][17];
  __shared__ float sVec[4][16];
  __shared__ float sU[4][2][16];
  __shared__ int   sNbr[4][28];

  const int wav  = threadIdx.x >> 5;
  const int lane = threadIdx.x & 31;
  const int m    = lane & 15;     // feature dimension owned by this lane
  const int lh   = lane >> 4;
  const int row  = blockIdx.x * 4 + wav;

  if (lane < 28) sNbr[wav][lane] = (lane < KNN) ? nbr[(size_t)row * KNN + lane] : 0;
  __syncthreads();

  for (int s = 0; s < 2; ++s) {
    const float* src = (s == 0) ? latent : raw;

    // Gather this lane's 14 k-slots (k = 4*kc + 2*lh + t); zeros for k>=25.
    float vals[14];
    float psum = 0.0f;
    #pragma unroll
    for (int kc = 0; kc < 7; ++kc) {
      #pragma unroll
      for (int t = 0; t < 2; ++t) {
        const int k = 4 * kc + 2 * lh + t;
        float x = 0.0f;
        if (k < KNN) x = src[(size_t)sNbr[wav][k] * DIMS + m];
        vals[kc * 2 + t] = x;
        psum += x;
      }
    }
    // Two lanes (l, l^16) share feature m and jointly cover all 25 k's.
    const float mean = (psum + __shfl_xor(psum, 16)) * (1.0f / 25.0f);

    v8f accv = {};
    #pragma unroll
    for (int kc = 0; kc < 7; ++kc) {
      const int k0 = 4 * kc + 2 * lh;
      v2f ab;
      ab.x = (k0     < KNN) ? vals[kc * 2]     - mean : 0.0f;
      ab.y = (k0 + 1 < KNN) ? vals[kc * 2 + 1] - mean : 0.0f;
      accv = wmma_f32_4(ab, ab, accv);           // cov (unscaled; eigvec invariant)
    }
    #pragma unroll
    for (int v = 0; v < 8; ++v)
      sCov[wav][v + 8 * lh][m] = accv[v];

    if (lane < 16) sVec[wav][lane] = 1.0f;
    __syncthreads();

    for (int it = 0; it < 24; ++it) {            // power iteration
      float y = 0.0f;
      if (lane < 16) {
        #pragma unroll
        for (int n = 0; n < 16; ++n) y += sCov[wav][lane][n] * sVec[wav][n];
      }
      __syncthreads();
      float n2 = y * y;                          // reduce within 16-lane half
      n2 += __shfl_xor(n2, 1); n2 += __shfl_xor(n2, 2);
      n2 += __shfl_xor(n2, 4); n2 += __shfl_xor(n2, 8);
      const float inv = rsqrtf(n2 + 1e-30f);
      if (lane < 16) sVec[wav][lane] = y * inv;
      __syncthreads();
    }
    if (lane < 16) sU[wav][s][lane] = sVec[wav][lane];
    __syncthreads();
  }

  float p = 0.0f;
  if (lane < 16) p = sU[wav][0][lane] * sU[wav][1][lane];
  p += __shfl_xor(p, 1); p += __shfl_xor(p, 2);
  p += __shfl_xor(p, 4); p += __shfl_xor(p, 8);
  if (lane == 0) atomicAdd(accTsa, 2.0f - 2.0f * p * p);   // ||Pz-Px||_F^2, rank-1
}

// ---------------------------------------------------------------------------
__global__ void finalize_kernel(const float* __restrict__ ws,
                                float* __restrict__ out) {
  if (threadIdx.x == 0)
    out[0] = ws[0] * (1.0f / ((float)B_PTS * (float)DIMS))
           + 0.1f * ws[1] * (1.0f / (float)B_PTS);
}

// ---------------------------------------------------------------------------
extern "C" void kernel_launch(void* const* d_in, const int* in_sizes, int n_in,
                              void* d_out, int out_size, void* d_ws, size_t ws_size,
                              hipStream_t stream) {
  const float* outputs = (const float*)d_in[0];
  const float* targets = (const float*)d_in[1];
  const float* latent  = (const float*)d_in[2];
  const float* raw     = (const float*)d_in[3];
  float* out = (float*)d_out;

  float* wsf = (float*)d_ws;              // [0]=recon accum, [1]=tsa accum
  int*   nbr = (int*)(wsf + 16);          // 16384*25 ints (~1.6 MB)

  zero_accum_kernel<<<1, 32, 0, stream>>>(wsf);
  recon_kernel<<<256, 256, 0, stream>>>(outputs, targets, wsf + 0, B_PTS * DIMS);
  knn_kernel<<<B_PTS / 16, 256, 0, stream>>>(raw, nbr);
  tsa_kernel<<<B_PTS / 4, 128, 0, stream>>>(latent, raw, nbr, wsf + 1);
  finalize_kernel<<<1, 32, 0, stream>>>(wsf, out);
}